// GELU232_23648089932103
// MI455X (gfx1250) — compile-verified
//
#include <hip/hip_runtime.h>
#include <hip/hip_bf16.h>
#include <math.h>

typedef __attribute__((ext_vector_type(2))) float v2f;
typedef __attribute__((ext_vector_type(8))) float v8f;

#define Bn 8
#define Tn 2048
#define Dn 2048
#define Nn 512
#define BTn (Bn * Tn)
#define NPART 128
#define EPSF 1e-12f

// workspace layout (float indices)
#define WS_MCURR 0          // D floats: column mean, then normalized in-place
#define WS_NV    2048       // D floats: normalized nearest vector
#define WS_PACK  4096       // 8-byte aligned slot: packed argmax (u64)
#define WS_PART  4352       // NPART * D floats: column-sum partials

__device__ __forceinline__ float gelu_f(float x) {
  const float c = 0.7978845608028654f; // sqrt(2/pi)
  return 0.5f * x * (1.0f + tanhf(c * (x + 0.044715f * x * x * x)));
}
__device__ __forceinline__ float sigmoid_f(float z) { return 1.0f / (1.0f + expf(-z)); }

__device__ float blockReduceSum(float v, float* red) {
  int tid = threadIdx.x;
  red[tid] = v;
  __syncthreads();
  for (int s = blockDim.x >> 1; s > 0; s >>= 1) {
    if (tid < s) red[tid] += red[tid + s];
    __syncthreads();
  }
  float r = red[0];
  __syncthreads();
  return r;
}

// Pass 1: y = gelu(x) -> out; per-block column partial sums (fixed order, deterministic).
// grid 128 blocks x 256 threads; block b handles tokens [b*128, b*128+128), all D columns.
__global__ void k_gelu_stats(const float* __restrict__ x, float* __restrict__ y,
                             float* __restrict__ ws) {
  const int tid = threadIdx.x;
  const int t0 = blockIdx.x * (BTn / NPART);
  float acc[8];
#pragma unroll
  for (int k = 0; k < 8; ++k) acc[k] = 0.0f;

  for (int t = t0; t < t0 + (BTn / NPART); ++t) {
    const long long base = (long long)t * Dn;
    __builtin_prefetch(&x[base + 2 * Dn + tid], 0, 1); // global_prefetch_b8
#pragma unroll
    for (int k = 0; k < 8; ++k) {
      long long idx = base + tid + 256 * k;
      float v = gelu_f(x[idx]);
      y[idx] = v;
      acc[k] += v;
    }
  }
#pragma unroll
  for (int k = 0; k < 8; ++k)
    ws[WS_PART + (long long)blockIdx.x * Dn + tid + 256 * k] = acc[k];
}

// Pass 2: m_curr[d] = sum_p part[p][d] / BT.  grid 8 x 256.
__global__ void k_colreduce(float* __restrict__ ws) {
  int d = blockIdx.x * 256 + threadIdx.x;
  float s = 0.0f;
  for (int p = 0; p < NPART; ++p) s += ws[WS_PART + (long long)p * Dn + d];
  ws[WS_MCURR + d] = s * (1.0f / (float)BTn);
}

// Pass 3: normalize m_curr in place; init argmax slot.  1 block x 256.
__global__ void k_mnorm(float* __restrict__ ws) {
  __shared__ float red[256];
  int tid = threadIdx.x;
  float ssq = 0.0f;
#pragma unroll
  for (int k = 0; k < 8; ++k) {
    float v = ws[WS_MCURR + tid + 256 * k];
    ssq += v * v;
  }
  float tot = blockReduceSum(ssq, red);
  float rinv = 1.0f / fmaxf(sqrtf(tot), EPSF);
#pragma unroll
  for (int k = 0; k < 8; ++k) ws[WS_MCURR + tid + 256 * k] *= rinv;
  if (tid == 0) *(unsigned long long*)&ws[WS_PACK] = 0ull;
}

// Pass 4: cosine sims + argmax via integer atomicMax on order-preserving key.
// grid 512 (one block per buffer row) x 256.
__global__ void k_nn(const float* __restrict__ buf, const unsigned char* __restrict__ mask,
                     float* __restrict__ ws) {
  __shared__ float red[256];
  int n = blockIdx.x, tid = threadIdx.x;
  float dot = 0.0f, ssq = 0.0f;
#pragma unroll
  for (int k = 0; k < 8; ++k) {
    float b = buf[(long long)n * Dn + tid + 256 * k];
    dot += b * ws[WS_MCURR + tid + 256 * k]; // m_curr already normalized
    ssq += b * b;
  }
  float dtot = blockReduceSum(dot, red);
  float stot = blockReduceSum(ssq, red);
  if (tid == 0) {
    float sim = dtot / fmaxf(sqrtf(stot), EPSF);
    if (!mask[n]) sim = -1.0f;
    unsigned u = __float_as_uint(sim);
    u = (u & 0x80000000u) ? ~u : (u | 0x80000000u); // order-preserving
    unsigned long long packed = ((unsigned long long)u << 32) | (unsigned)(~(unsigned)n);
    atomicMax((unsigned long long*)&ws[WS_PACK], packed); // ties -> smallest n
  }
}

// Pass 5: nv_n = normalize(buf[best]).  1 block x 256.
__global__ void k_nvprep(const float* __restrict__ buf, float* __restrict__ ws) {
  __shared__ float red[256];
  int tid = threadIdx.x;
  unsigned long long packed = *(unsigned long long*)&ws[WS_PACK];
  int best = (int)(~(unsigned)(packed & 0xFFFFFFFFull));
  float ssq = 0.0f;
#pragma unroll
  for (int k = 0; k < 8; ++k) {
    float v = buf[(long long)best * Dn + tid + 256 * k];
    ssq += v * v;
  }
  float tot = blockReduceSum(ssq, red);
  float rinv = 1.0f / fmaxf(sqrtf(tot), EPSF);
#pragma unroll
  for (int k = 0; k < 8; ++k)
    ws[WS_NV + tid + 256 * k] = buf[(long long)best * Dn + tid + 256 * k] * rinv;
}

// Pass 6: fused gate + scale. One wave (32 threads) per 16 tokens; LDS-staged tiles
// feed V_WMMA_F32_16X16X4_F32 with B replicated across all 16 columns so every
// D column holds the same dot.  grid 1024 x 32.
#define KT 256
#define PITCH 257
__global__ void k_gate_wmma(float* __restrict__ y, const float* __restrict__ ws,
                            const float* __restrict__ p_ls, const float* __restrict__ p_lt,
                            const float* __restrict__ p_lg) {
  __shared__ float tileY[16 * PITCH];
  __shared__ float tileNV[KT];
  __shared__ float sh_dots[16];
  __shared__ float sh_gate[16];

  const int lane = threadIdx.x; // blockDim.x == 32
  const int t0 = blockIdx.x * 16;
  const int m = lane & 15;
  const int koff = (lane < 16) ? 0 : 2;

  v8f c = {};
  float rowssq = 0.0f; // lanes 0..15: sum y^2 over full D for token (t0+lane)

  for (int kb = 0; kb < Dn; kb += KT) {
    // cooperative coalesced load: 16 rows x 256 cols of y, plus nv chunk
#pragma unroll 4
    for (int i = 0; i < (16 * KT) / 32; ++i) {
      int e = lane + 32 * i;
      int row = e >> 8, col = e & (KT - 1);
      tileY[row * PITCH + col] = y[(long long)(t0 + row) * Dn + kb + col];
    }
#pragma unroll
    for (int i = 0; i < KT / 32; ++i) {
      int e = lane + 32 * i;
      tileNV[e] = ws[WS_NV + kb + e];
    }
    __syncthreads();

    // A: 16x4 f32 (lane<16: K=k0,k0+1; lane>=16: K=k0+2,k0+3); B: all cols = nv chunk
    for (int k0 = 0; k0 < KT; k0 += 4) {
      v2f a, b;
      a.x = tileY[m * PITCH + k0 + koff];
      a.y = tileY[m * PITCH + k0 + koff + 1];
      b.x = tileNV[k0 + koff];
      b.y = tileNV[k0 + koff + 1];
      c = __builtin_amdgcn_wmma_f32_16x16x4_f32(false, a, false, b, (short)0, c,
                                                false, false);
    }

    if (lane < 16) { // per-token ||y||^2 from the staged tile (conflict-free: pitch 257)
      float s = 0.0f;
      for (int col = 0; col < KT; ++col) {
        float v = tileY[lane * PITCH + col];
        s += v * v;
      }
      rowssq += s;
    }
    __syncthreads();
  }

  // C/D layout: VGPR v -> M=v (lanes 0-15), M=v+8 (lanes 16-31); all N equal.
  if (lane == 0) {
#pragma unroll
    for (int v = 0; v < 8; ++v) sh_dots[v] = c[v];
  }
  if (lane == 16) {
#pragma unroll
    for (int v = 0; v < 8; ++v) sh_dots[8 + v] = c[v];
  }
  __syncthreads();

  if (lane < 16) {
    float sharp = fminf(fmaxf(expf(p_ls[0]), 1.0f), 20.0f);
    float thr = sigmoid_f(p_lt[0]);
    float gmin = 0.05f + 0.45f * sigmoid_f(p_lg[0]);
    float ts = sh_dots[lane] / fmaxf(sqrtf(rowssq), EPSF);
    float g = gmin + (1.0f - gmin) * sigmoid_f(-sharp * (ts - thr));
    sh_gate[lane] = g;
  }
  __syncthreads();

  // scale in place: out = y * gate (vectorized b128 traffic)
  for (int row = 0; row < 16; ++row) {
    float g = sh_gate[row];
    float4* p = (float4*)&y[(long long)(t0 + row) * Dn];
#pragma unroll
    for (int j = 0; j < Dn / (32 * 4); ++j) {
      float4 v = p[lane + 32 * j];
      v.x *= g; v.y *= g; v.z *= g; v.w *= g;
      p[lane + 32 * j] = v;
    }
  }
}

extern "C" void kernel_launch(void* const* d_in, const int* in_sizes, int n_in,
                              void* d_out, int out_size, void* d_ws, size_t ws_size,
                              hipStream_t stream) {
  const float* x = (const float*)d_in[0];
  const float* buf = (const float*)d_in[1];
  const unsigned char* mask = (const unsigned char*)d_in[2];
  const float* ls = (const float*)d_in[3];
  const float* lt = (const float*)d_in[4];
  const float* lg = (const float*)d_in[5];
  float* out = (float*)d_out;
  float* ws = (float*)d_ws;

  k_gelu_stats<<<NPART, 256, 0, stream>>>(x, out, ws);
  k_colreduce<<<Dn / 256, 256, 0, stream>>>(ws);
  k_mnorm<<<1, 256, 0, stream>>>(ws);
  k_nn<<<Nn, 256, 0, stream>>>(buf, mask, ws);
  k_nvprep<<<1, 256, 0, stream>>>(buf, ws);
  k_gate_wmma<<<BTn / 16, 32, 0, stream>>>(out, ws, ls, lt, lg);
}